// AttentionBase_85839216377907
// MI455X (gfx1250) — compile-verified
//
#include <hip/hip_runtime.h>

typedef __attribute__((ext_vector_type(2))) float v2f;
typedef __attribute__((ext_vector_type(8))) float v8f;

#define WAVES   8
#define DDIM    1024
#define QDIM    1024
#define KDIM    1024
#define DSLICE  128   // D-contraction slice per wave (QK^T)
#define DVSLICE 128   // Dv output slice per wave (PV)
#define NEGINF  (-3.0e38f)

__global__ __launch_bounds__(256) void attn_flash_f32_wmma(
    const float* __restrict__ Qm, const float* __restrict__ Km,
    const float* __restrict__ Vm, const int* __restrict__ qlens,
    const int* __restrict__ klens, float* __restrict__ Om)
{
    __shared__ float sPart[WAVES * 256];  // per-wave partial score tiles
    __shared__ float sS[256];             // reduced scores / P tile (16x16)
    __shared__ float sM[16], sL[16], sAlpha[16];

    const int b    = blockIdx.x >> 6;     // 64 q-tiles per batch
    const int q0   = (blockIdx.x & 63) * 16;
    const int tid  = threadIdx.x;
    const int wave = tid >> 5;
    const int lane = tid & 31;
    const int l15  = lane & 15;
    const int hi2  = (lane >> 4) * 2;     // K-offset within WMMA step for hi half-wave
    const int hi8  = (lane >> 4) * 8;     // row offset for C/D layout hi half-wave

    const int ql  = qlens[b];
    const int kl  = klens[b];
    const int nkt = (kl + 15) >> 4;       // key tiles actually needed (mask-aware)

    const float* Qb = Qm + (size_t)b * QDIM * DDIM + (size_t)q0 * DDIM;
    const float* Kb = Km + (size_t)b * KDIM * DDIM;
    const float* Vb = Vm + (size_t)b * KDIM * DDIM;
    float*       Ob = Om + (size_t)b * QDIM * DDIM + (size_t)q0 * DDIM;

    if (tid < 16) { sM[tid] = NEGINF; sL[tid] = 0.0f; }

    // Register-resident Q slice for this wave in WMMA f32 16x4 A-layout:
    // step ds: a = { Q[row=l15][dbase + 4*ds + hi2], Q[row][dbase + 4*ds + hi2 + 1] }
    const int dbase = wave * DSLICE;
    v2f qreg[32];
#pragma unroll
    for (int ds = 0; ds < 32; ++ds)
        qreg[ds] = *(const v2f*)(Qb + (size_t)l15 * DDIM + dbase + ds * 4 + hi2);

    v8f oacc[8];
#pragma unroll
    for (int t = 0; t < 8; ++t) { v8f z = {}; oacc[t] = z; }

    __syncthreads();

    for (int kt = 0; kt < nkt; ++kt) {
        // ---- partial S = Q[16 x 128] * K^T[128 x 16] over this wave's D-slice
        // Two independent accumulators break the WMMA->WMMA RAW chain on C.
        const float* Kt = Kb + (size_t)(kt * 16 + l15) * DDIM + dbase + hi2;
        __builtin_prefetch(Kt + (size_t)16 * DDIM, 0, 0);  // next K tile -> global_prefetch_b8
        v8f s0 = {}, s1 = {};
#pragma unroll
        for (int ds = 0; ds < 32; ds += 2) {
            v2f bk0 = *(const v2f*)(Kt + ds * 4);
            v2f bk1 = *(const v2f*)(Kt + ds * 4 + 4);
            s0 = __builtin_amdgcn_wmma_f32_16x16x4_f32(
                    false, qreg[ds],     false, bk0, (short)0, s0, false, false);
            s1 = __builtin_amdgcn_wmma_f32_16x16x4_f32(
                    false, qreg[ds + 1], false, bk1, (short)0, s1, false, false);
        }
        // spill partial tile to LDS in C/D layout: vgpr r, lane -> row r+hi8, col l15
        {
            float* sp = sPart + wave * 256;
#pragma unroll
            for (int r = 0; r < 8; ++r)
                sp[(r + hi8) * 16 + l15] = s0[r] + s1[r];
        }
        __syncthreads();

        // ---- cross-wave reduction + key-length mask (one element per thread)
        {
            float acc = 0.0f;
#pragma unroll
            for (int w = 0; w < WAVES; ++w) acc += sPart[w * 256 + tid];
            const int kglob = kt * 16 + (tid & 15);
            sS[tid] = (kglob < kl) ? acc : NEGINF;
        }
        __syncthreads();

        // ---- online softmax per row (16 threads, one row each)
        if (tid < 16) {
            float mold = sM[tid];
            float mnew = mold;
#pragma unroll
            for (int c = 0; c < 16; ++c) mnew = fmaxf(mnew, sS[tid * 16 + c]);
            const float alpha = __expf(mold - mnew);
            float lsum = 0.0f;
#pragma unroll
            for (int c = 0; c < 16; ++c) {
                const float p = __expf(sS[tid * 16 + c] - mnew);
                sS[tid * 16 + c] = p;
                lsum += p;
            }
            sM[tid]     = mnew;
            sL[tid]     = sL[tid] * alpha + lsum;
            sAlpha[tid] = alpha;
        }
        __syncthreads();

        // ---- O = O*alpha + P[16x16] * V[16 x 128]  (this wave's Dv slice)
        float alf[8];
#pragma unroll
        for (int r = 0; r < 8; ++r) alf[r] = sAlpha[r + hi8];

        const int dvbase = wave * DVSLICE;
#pragma unroll
        for (int t = 0; t < 8; ++t) {
            v8f o = oacc[t];
#pragma unroll
            for (int r = 0; r < 8; ++r) o[r] *= alf[r];
            const float* Vt = Vb + dvbase + t * 16 + l15;
            __builtin_prefetch(Vt + (size_t)16 * DDIM, 0, 0);  // V row 16 (key tile 1), same for every kt
#pragma unroll
            for (int st = 0; st < 4; ++st) {
                v2f a = *(const v2f*)(sS + l15 * 16 + st * 4 + hi2);  // P in A-layout
                const int kk = kt * 16 + st * 4 + hi2;
                v2f bb;
                bb.x = Vt[(size_t)kk * DDIM];
                bb.y = Vt[(size_t)(kk + 1) * DDIM];
                o = __builtin_amdgcn_wmma_f32_16x16x4_f32(
                        false, a, false, bb, (short)0, o, false, false);
            }
            oacc[t] = o;
        }
        __syncthreads();   // protect sS / sPart for next iteration
    }

    // ---- epilogue: divide by l, zero masked q rows, store
    float linv[8];
#pragma unroll
    for (int r = 0; r < 8; ++r) {
        const float lv = sL[r + hi8];
        linv[r] = (lv > 0.0f) ? (1.0f / lv) : 0.0f;
    }
    const int dvbase = wave * DVSLICE;
#pragma unroll
    for (int t = 0; t < 8; ++t) {
#pragma unroll
        for (int r = 0; r < 8; ++r) {
            const int row = r + hi8;
            const float val = (q0 + row < ql) ? oacc[t][r] * linv[r] : 0.0f;
            Ob[(size_t)row * DDIM + dvbase + t * 16 + l15] = val;
        }
    }
}

extern "C" void kernel_launch(void* const* d_in, const int* in_sizes, int n_in,
                              void* d_out, int out_size, void* d_ws, size_t ws_size,
                              hipStream_t stream) {
    (void)in_sizes; (void)n_in; (void)d_ws; (void)ws_size; (void)out_size;
    const float* Q  = (const float*)d_in[0];
    const float* K  = (const float*)d_in[1];
    const float* V  = (const float*)d_in[2];
    const int*   ql = (const int*)d_in[3];
    const int*   kl = (const int*)d_in[4];
    float*       O  = (float*)d_out;

    dim3 grid(16 * (QDIM / 16));   // B * q-tiles = 1024 workgroups
    dim3 block(256);               // 8 waves (wave32)
    hipLaunchKernelGGL(attn_flash_f32_wmma, grid, block, 0, stream,
                       Q, K, V, ql, kl, O);
}